// DGI_81698867904739
// MI455X (gfx1250) — compile-verified
//
#include <hip/hip_runtime.h>

#define NN   50000      // nodes (== 3125 * 16, exact tiling)
#define NE   800000     // edges
#define HD   128        // feature dim (F_IN == H == 128)

typedef float v2f __attribute__((ext_vector_type(2)));
typedef float v8f __attribute__((ext_vector_type(8)));

// ---------------------------------------------------------------------------
// Degree / norm kernels
// ---------------------------------------------------------------------------
__global__ __launch_bounds__(256) void dgi_deg_init(float* __restrict__ deg) {
    int i = blockIdx.x * blockDim.x + threadIdx.x;
    if (i < NN) deg[i] = 1.0f;                 // self-loop contribution
}

__global__ __launch_bounds__(256) void dgi_deg_accum(const int* __restrict__ dst,
                                                     float* __restrict__ deg) {
    int e = blockIdx.x * blockDim.x + threadIdx.x;
    if (e < NE) atomicAdd(&deg[dst[e]], 1.0f);
}

__global__ __launch_bounds__(256) void dgi_dinv(float* __restrict__ degdinv) {
    int i = blockIdx.x * blockDim.x + threadIdx.x;
    if (i < NN) {
        float d = degdinv[i];
        degdinv[i] = (d > 0.0f) ? rsqrtf(d) : 0.0f;
    }
}

// ---------------------------------------------------------------------------
// f32 WMMA GEMM:  C[M,128] = A[M,128] @ B[128,128]
// One wave per 16x16 output tile; 8 waves/block cover the 128 columns.
// grid.x = M/16 (M == NN == 3125*16).
// A-frag (16x4 f32): lane l, half h -> A[l][2h+i] in vgpr i
// B-frag (4x16 f32): lane l, half h -> B[2h+i][l] in vgpr i
// D (16x16 f32):     lane l, half h, vgpr r -> D[r+8h][l]
// ---------------------------------------------------------------------------
__global__ __launch_bounds__(256) void dgi_gemm128_wmma(const float* __restrict__ A,
                                                        const float* __restrict__ B,
                                                        float* __restrict__ C) {
    const int wave = threadIdx.x >> 5;         // 0..7 -> column tile
    const int lane = threadIdx.x & 31;
    const int half = lane >> 4;                // 0 or 1
    const int l    = lane & 15;
    const int m0   = blockIdx.x * 16;
    const int n0   = wave * 16;

    const float* __restrict__ arow = A + (size_t)(m0 + l) * HD;

    v8f acc = {0.f, 0.f, 0.f, 0.f, 0.f, 0.f, 0.f, 0.f};

#pragma unroll
    for (int k0 = 0; k0 < HD; k0 += 4) {
        const int ka = k0 + 2 * half;
        v2f a, b;
        a.x = arow[ka + 0];
        a.y = arow[ka + 1];
        b.x = B[(size_t)(ka + 0) * HD + n0 + l];
        b.y = B[(size_t)(ka + 1) * HD + n0 + l];
        // (neg_a, A, neg_b, B, c_mod, C, reuse_a, reuse_b)
        acc = __builtin_amdgcn_wmma_f32_16x16x4_f32(
            false, a, false, b, (short)0, acc, false, false);
    }

#pragma unroll
    for (int r = 0; r < 8; ++r) {
        C[(size_t)(m0 + r + 8 * half) * HD + n0 + l] = acc[r];
    }
}

// ---------------------------------------------------------------------------
// z init: z[n,c] = bias[c] + dinv[n]^2 * xw[n,c]   (self-loop message)
// ---------------------------------------------------------------------------
__global__ __launch_bounds__(256) void dgi_z_init(const float* __restrict__ xw,
                                                  const float* __restrict__ dinv,
                                                  const float* __restrict__ bias,
                                                  float* __restrict__ z) {
    long long i = (long long)blockIdx.x * blockDim.x + threadIdx.x;
    if (i < (long long)NN * HD) {
        int n  = (int)(i >> 7);
        int ch = (int)(i & 127);
        float di = dinv[n];
        z[i] = bias[ch] + di * di * xw[i];
    }
}

// ---------------------------------------------------------------------------
// Edge scatter: z[dst] += dinv[src]*dinv[dst] * xw[src]
// One thread per (edge, channel): lanes within a wave share the edge, so the
// gather read and the atomic scatter are both contiguous 128B bursts.
// ---------------------------------------------------------------------------
__global__ __launch_bounds__(256) void dgi_edge_scatter(const int* __restrict__ src,
                                                        const int* __restrict__ dst,
                                                        const float* __restrict__ xw,
                                                        const float* __restrict__ dinv,
                                                        float* __restrict__ z) {
    long long i = (long long)blockIdx.x * blockDim.x + threadIdx.x;
    if (i < (long long)NE * HD) {
        int e  = (int)(i >> 7);
        int ch = (int)(i & 127);
        int s  = src[e];
        int d  = dst[e];
        float c = dinv[s] * dinv[d];
        atomicAdd(&z[(long long)d * HD + ch], xw[(long long)s * HD + ch] * c);
    }
}

// ---------------------------------------------------------------------------
// Scores: pos[n] = u[n].z[n] + bb ; neg[n] = u[n].z[perm[n]] + bb
// One wave32 per node; float4 per lane; shfl_xor tree reduction.
// ---------------------------------------------------------------------------
__global__ __launch_bounds__(256) void dgi_score(const float* __restrict__ u,
                                                 const float* __restrict__ z,
                                                 const int* __restrict__ perm,
                                                 const float* __restrict__ bb,
                                                 float* __restrict__ out) {
    const int wave = threadIdx.x >> 5;
    const int lane = threadIdx.x & 31;
    const int n = blockIdx.x * 8 + wave;
    if (n >= NN) return;

    const float* un = u + (size_t)n * HD;
    const float* za = z + (size_t)n * HD;
    const float* zb = z + (size_t)perm[n] * HD;

    float4 uu = *(const float4*)(un + lane * 4);
    float4 va = *(const float4*)(za + lane * 4);
    float4 vb = *(const float4*)(zb + lane * 4);

    float pos = uu.x * va.x + uu.y * va.y + uu.z * va.z + uu.w * va.w;
    float neg = uu.x * vb.x + uu.y * vb.y + uu.z * vb.z + uu.w * vb.w;

#pragma unroll
    for (int off = 16; off > 0; off >>= 1) {
        pos += __shfl_xor(pos, off, 32);
        neg += __shfl_xor(neg, off, 32);
    }
    if (lane == 0) {
        float bias = bb[0];
        out[n]      = pos + bias;
        out[NN + n] = neg + bias;
    }
}

// ---------------------------------------------------------------------------
// Launch
// ---------------------------------------------------------------------------
extern "C" void kernel_launch(void* const* d_in, const int* in_sizes, int n_in,
                              void* d_out, int out_size, void* d_ws, size_t ws_size,
                              hipStream_t stream) {
    (void)in_sizes; (void)n_in; (void)out_size; (void)ws_size;

    const float* x    = (const float*)d_in[0];   // [N, 128]
    const float* W    = (const float*)d_in[1];   // [128, 128]
    const float* b    = (const float*)d_in[2];   // [128]
    const float* Wb   = (const float*)d_in[3];   // [1, 128, 128]
    const float* bb   = (const float*)d_in[4];   // [1]
    const int*   ei   = (const int*)d_in[5];     // [2, E]
    const int*   perm = (const int*)d_in[6];     // [N]
    const int*   esrc = ei;
    const int*   edst = ei + NE;
    float*       out  = (float*)d_out;           // [N] pos ++ [N] neg

    // Workspace layout (256B aligned chunks)
    char* ws = (char*)d_ws;
    size_t off = 0;
    float* dinv = (float*)(ws + off); off += ((size_t)NN * 4 + 255) & ~(size_t)255;
    float* xw   = (float*)(ws + off); off += ((size_t)NN * HD * 4 + 255) & ~(size_t)255;
    float* z    = (float*)(ws + off); off += ((size_t)NN * HD * 4 + 255) & ~(size_t)255;
    float* u    = (float*)(ws + off); off += ((size_t)NN * HD * 4 + 255) & ~(size_t)255;

    const int nblk  = (NN + 255) / 256;
    const int eblk  = (NE + 255) / 256;
    const int nhblk = (int)(((long long)NN * HD + 255) / 256);
    const int ehblk = (int)(((long long)NE * HD + 255) / 256);

    // 1) symmetric-norm coefficients
    dgi_deg_init<<<nblk, 256, 0, stream>>>(dinv);
    dgi_deg_accum<<<eblk, 256, 0, stream>>>(edst, dinv);
    dgi_dinv<<<nblk, 256, 0, stream>>>(dinv);

    // 2) xw = x @ W   (f32 WMMA)
    dgi_gemm128_wmma<<<NN / 16, 256, 0, stream>>>(x, W, xw);

    // 3) z = segment_sum(norm messages) + b
    dgi_z_init<<<nhblk, 256, 0, stream>>>(xw, dinv, b, z);
    dgi_edge_scatter<<<ehblk, 256, 0, stream>>>(esrc, edst, xw, dinv, z);

    // 4) u = z @ Wb[0]  (f32 WMMA)
    dgi_gemm128_wmma<<<NN / 16, 256, 0, stream>>>(z, Wb, u);

    // 5) bilinear scores
    dgi_score<<<(NN + 7) / 8, 256, 0, stream>>>(u, z, perm, bb, out);
}